// GCN_64201171140671
// MI455X (gfx1250) — compile-verified
//
#include <hip/hip_runtime.h>
#include <hip/hip_bf16.h>

// CDNA5 / gfx1250 GCN forward pass.
// Pipeline: l2norm -> [WMMA f32 GEMM -> COO scatter-add (L2 atomics)] x3.
// All WMMA tiles are exact (N=100000 and 128/64/32 are multiples of 16),
// so EXEC is all-ones at every v_wmma as the ISA requires.

typedef __attribute__((ext_vector_type(2))) float v2f;
typedef __attribute__((ext_vector_type(8))) float v8f;

// ---------------------------------------------------------------------------
// L2 row-normalize: one wave (32 lanes) per 256-float row.
// ---------------------------------------------------------------------------
__global__ __launch_bounds__(256) void l2norm_rows(const float* __restrict__ F,
                                                   float* __restrict__ X,
                                                   int nRows) {
    const int lane = threadIdx.x & 31;
    const int row  = blockIdx.x * 8 + (threadIdx.x >> 5);
    if (row >= nRows) return;
    const float4* f = (const float4*)(F + (size_t)row * 256);
    float4 v0 = f[lane];        // floats [4*lane     .. 4*lane+3]
    float4 v1 = f[lane + 32];   // floats [128+4*lane .. ]
    float s = v0.x*v0.x + v0.y*v0.y + v0.z*v0.z + v0.w*v0.w
            + v1.x*v1.x + v1.y*v1.y + v1.z*v1.z + v1.w*v1.w;
    #pragma unroll
    for (int off = 16; off > 0; off >>= 1) s += __shfl_xor(s, off, 32);
    const float inv = rsqrtf(fmaxf(s, 1e-12f));
    v0.x *= inv; v0.y *= inv; v0.z *= inv; v0.w *= inv;
    v1.x *= inv; v1.y *= inv; v1.z *= inv; v1.w *= inv;
    float4* xo = (float4*)(X + (size_t)row * 256);
    xo[lane]      = v0;
    xo[lane + 32] = v1;
}

// ---------------------------------------------------------------------------
// Y[nRows x M] = (RELU? max(X,0):X)[nRows x K] @ W[K x M], row-major.
// One wave computes one 16x16 tile via V_WMMA_F32_16X16X4_F32.
// A 16x4 f32 frag: lanes 0-15 hold K=k,k+1 (vgpr0,1); lanes 16-31 K=k+2,k+3.
// B 4x16 f32 frag mirrored: lanes 0-15 rows K=k,k+1; lanes 16-31 K=k+2,k+3.
// C/D: vgpr r -> lanes 0-15: (M=r, N=lane), lanes 16-31: (M=r+8, N=lane-16).
// ---------------------------------------------------------------------------
template <int K, int M, bool RELU>
__global__ __launch_bounds__(128) void gemm_wmma_f32(const float* __restrict__ X,
                                                     const float* __restrict__ W,
                                                     float* __restrict__ Y,
                                                     int nRows) {
    const int lane   = threadIdx.x & 31;
    const int waveId = blockIdx.x * 4 + (threadIdx.x >> 5);
    const int tilesN = M / 16;
    const int totalTiles = (nRows / 16) * tilesN;
    if (waveId >= totalTiles) return;           // wave-uniform exit
    const int rowTile = waveId / tilesN;
    const int colTile = waveId % tilesN;

    const int halfSel = lane >> 4;              // 0: lanes 0-15, 1: lanes 16-31
    const int kOff    = halfSel * 2;            // K sub-offset for A/B frags
    const int aRow    = rowTile * 16 + (lane & 15);
    const int bCol    = colTile * 16 + (lane & 15);

    const float* __restrict__ xrow = X + (size_t)aRow * K;
    v8f c = {};
    #pragma unroll 4
    for (int k = 0; k < K; k += 4) {
        float a0 = xrow[k + kOff];
        float a1 = xrow[k + kOff + 1];
        if (RELU) { a0 = fmaxf(a0, 0.0f); a1 = fmaxf(a1, 0.0f); }
        v2f a; a.x = a0; a.y = a1;
        v2f b;
        b.x = W[(size_t)(k + kOff)     * M + bCol];
        b.y = W[(size_t)(k + kOff + 1) * M + bCol];
        c = __builtin_amdgcn_wmma_f32_16x16x4_f32(
                false, a, false, b, (short)0, c, false, false);
    }
    const int mBase = rowTile * 16 + halfSel * 8;
    #pragma unroll
    for (int r = 0; r < 8; ++r)
        Y[(size_t)(mBase + r) * M + bCol] = c[r];
}

// ---------------------------------------------------------------------------
// COO SpMM scatter: out[row[e], :] += val[e] * H[col[e], :]
// One wave per edge. Each lane owns a contiguous D/32-float chunk of the row:
// the gather is one global_load_b128/b64/b32 per lane, the scatter is D/32
// contiguous global_atomic_add_f32 per lane (resolve in L2; all dense
// intermediates fit in the 192 MB L2).
// ---------------------------------------------------------------------------
template <int D>
__global__ __launch_bounds__(256) void spmm_scatter(const float* __restrict__ H,
                                                    const float* __restrict__ vals,
                                                    const int* __restrict__ rows,
                                                    const int* __restrict__ cols,
                                                    float* __restrict__ out,
                                                    int nEdges) {
    constexpr int PER = D / 32;                 // floats per lane (4 / 2 / 1)
    typedef __attribute__((ext_vector_type(PER))) float vPf;

    const int lane = threadIdx.x & 31;
    const int edge = blockIdx.x * 8 + (threadIdx.x >> 5);
    if (edge >= nEdges) return;
    const float v    = vals[edge];
    const int   r    = rows[edge];
    const int   cidx = cols[edge];

    const vPf src = *(const vPf*)(H + (size_t)cidx * D + lane * PER);
    float* __restrict__ dst = out + (size_t)r * D + lane * PER;
    #pragma unroll
    for (int j = 0; j < PER; ++j)
        atomicAdd(&dst[j], v * src[j]);
}

// ---------------------------------------------------------------------------
// Launch: inputs = {features, edge_vals, W0, W1, W2, edge_row, edge_col}
// ---------------------------------------------------------------------------
extern "C" void kernel_launch(void* const* d_in, const int* in_sizes, int n_in,
                              void* d_out, int out_size, void* d_ws, size_t ws_size,
                              hipStream_t stream) {
    const float* features = (const float*)d_in[0];
    const float* edgeVals = (const float*)d_in[1];
    const float* W0       = (const float*)d_in[2];
    const float* W1       = (const float*)d_in[3];
    const float* W2       = (const float*)d_in[4];
    const int*   edgeRow  = (const int*)d_in[5];
    const int*   edgeCol  = (const int*)d_in[6];

    const int N = in_sizes[0] / 256;   // 100000
    const int E = in_sizes[1];         // 1600000

    float* ws = (float*)d_ws;
    // Workspace layout with reuse (peak N*512 floats ~ 205 MB):
    float* xnorm = ws;                       // [N,256]; later reused as t1 [N,64]
    float* t0    = ws + (size_t)N * 256;     // [N,128]; later reused as g1 [N,64]
    float* g0    = ws + (size_t)N * 384;     // [N,128]; later reused as t2 [N,32]
    float* t1    = xnorm;
    float* g1    = t0;
    float* t2    = g0;
    float* out   = (float*)d_out;            // [N,32]

    const int normBlocks = (N + 7) / 8;
    const int spmmBlocks = (E + 7) / 8;
    auto gemmBlocks = [N](int m) { return ((N / 16) * (m / 16) + 3) / 4; };

    // x = l2_normalize(features)
    l2norm_rows<<<normBlocks, 256, 0, stream>>>(features, xnorm, N);

    // layer 1: g0 = spmm(x @ W0)   (relu deferred into next GEMM's A-load)
    gemm_wmma_f32<256, 128, false><<<gemmBlocks(128), 128, 0, stream>>>(xnorm, W0, t0, N);
    hipMemsetAsync(g0, 0, (size_t)N * 128 * sizeof(float), stream);
    spmm_scatter<128><<<spmmBlocks, 256, 0, stream>>>(t0, edgeVals, edgeRow, edgeCol, g0, E);

    // layer 2: g1 = spmm(relu(g0) @ W1)
    gemm_wmma_f32<128, 64, true><<<gemmBlocks(64), 128, 0, stream>>>(g0, W1, t1, N);
    hipMemsetAsync(g1, 0, (size_t)N * 64 * sizeof(float), stream);
    spmm_scatter<64><<<spmmBlocks, 256, 0, stream>>>(t1, edgeVals, edgeRow, edgeCol, g1, E);

    // layer 3: out = spmm(relu(g1) @ W2)
    gemm_wmma_f32<64, 32, true><<<gemmBlocks(32), 128, 0, stream>>>(g1, W2, t2, N);
    hipMemsetAsync(out, 0, (size_t)N * 32 * sizeof(float), stream);
    spmm_scatter<32><<<spmmBlocks, 256, 0, stream>>>(t2, edgeVals, edgeRow, edgeCol, out, E);
}